// HighPassFilter_27307402068460
// MI455X (gfx1250) — compile-verified
//
#include <hip/hip_runtime.h>
#include <math.h>

// Biquad high-pass IIR, [B=128,C=2,T=65536] f32. HBM-bound (144MB moved ~ 6.2us floor
// at 23.3 TB/s). Parallelized via chunked warm-up (stable filter, pole |z|=0.8234,
// 128-sample warmup -> state error ~1.6e-11). CDNA5 path: coalesced async global->LDS
// staging (global_load_async_to_lds_b128 + s_wait_asynccnt), double-buffered per wave.

#define T_LEN   65536
#define NCH     256
#define L_CHUNK 512           // stored samples per thread
#define W_WARM  128           // warm-up samples per thread (discarded)
#define ROWF    132           // 128 floats + 4 pad (row stride 132 % 64 banks = 4 -> 2-way max)
#define LDS_STAGE (32 * ROWF) // floats per stage buffer (one 128-sample slice per lane)

// one biquad step: y = b0*x + b1*x1 + b2*x2 - a1*y1 - a2*y2  (na1=-a1, na2=-a2)
#define BQ_STEP(xin, yout) do {                                   \
    float t_ = fmaf(b2, x2, fmaf(b1, x1, b0 * (xin)));            \
    float y_ = fmaf(na1, y1, fmaf(na2, y2, t_));                  \
    x2 = x1; x1 = (xin); y2 = y1; y1 = y_; (yout) = y_; } while (0)

__global__ __launch_bounds__(32)
void biquad_hp_kernel(const float* __restrict__ audio,
                      float* __restrict__ out,
                      float b0, float b1, float b2, float na1, float na2)
{
    __shared__ alignas(16) float smem[2 * LDS_STAGE];   // 33,792 B double buffer

    const int lane    = threadIdx.x;        // wave32: one wave per workgroup
    const int ch      = blockIdx.x >> 2;    // 4 waves per channel (128 chunks / 32 lanes)
    const int wchunk0 = (blockIdx.x & 3) * 32;
    const int k       = wchunk0 + lane;     // this lane's chunk index within channel

    const float* chbase = audio + (size_t)ch * T_LEN;
    const unsigned long long saddr = (unsigned long long)(uintptr_t)chbase;
    const int wbase = wchunk0 * L_CHUNK;    // wave's first stored sample in channel

    const unsigned lds_base0 = (unsigned)(uintptr_t)(&smem[0]); // low 32 bits = LDS offset

    // Stage s (0..4): 128-sample slice of every lane's [k*512-128, k*512+512) window.
    // Instruction j moves row j (lane k=wchunk0+j): 32 lanes x 16B contiguous = coalesced.
    auto issue_stage = [&](int s) {
        const unsigned ldsb = lds_base0 + (unsigned)((s & 1) * LDS_STAGE * 4);
        const int stage_off = (wbase + s * W_WARM - W_WARM) * 4;   // bytes
        #pragma unroll
        for (int j = 0; j < 32; ++j) {
            int start_b = stage_off + j * (L_CHUNK * 4);
            if (start_b < 0) start_b = 0;   // only channel-start warmup slice (unused data)
            const int voff = start_b + lane * 16;
            const unsigned lds_addr = ldsb + (unsigned)(j * ROWF * 4) + (unsigned)(lane * 16);
            asm volatile("global_load_async_to_lds_b128 %0, %1, %2"
                         :: "v"(lds_addr), "v"(voff), "s"(saddr)
                         : "memory");
        }
    };

    float x1 = 0.f, x2 = 0.f, y1 = 0.f, y2 = 0.f;
    const bool first = (k == 0);            // chunk 0: exact zero-state start, no warmup
    float4* outp = reinterpret_cast<float4*>(out + (size_t)ch * T_LEN + (size_t)k * L_CHUNK);

    issue_stage(0);
    for (int s = 0; s < 5; ++s) {
        if (s < 4) {
            issue_stage(s + 1);
            // async loads complete in order: <=32 outstanding => stage s fully landed
            asm volatile("s_wait_asynccnt 32" ::: "memory");
        } else {
            asm volatile("s_wait_asynccnt 0" ::: "memory");
        }

        const float4* row = reinterpret_cast<const float4*>(
            &smem[(size_t)((s & 1) * LDS_STAGE) + (size_t)lane * ROWF]);

        if (s == 0) {
            if (!first) {                    // warm-up: run filter, discard outputs
                #pragma unroll 4
                for (int i = 0; i < 32; ++i) {
                    float4 xv = row[i];
                    float dump;
                    BQ_STEP(xv.x, dump); BQ_STEP(xv.y, dump);
                    BQ_STEP(xv.z, dump); BQ_STEP(xv.w, dump);
                }
            }
        } else {
            float4* op = outp + (size_t)(s - 1) * 32;
            #pragma unroll 4
            for (int i = 0; i < 32; ++i) {
                float4 xv = row[i];
                float4 o;
                BQ_STEP(xv.x, o.x); BQ_STEP(xv.y, o.y);
                BQ_STEP(xv.z, o.z); BQ_STEP(xv.w, o.w);
                op[i] = o;
            }
        }
    }
}

extern "C" void kernel_launch(void* const* d_in, const int* in_sizes, int n_in,
                              void* d_out, int out_size, void* d_ws, size_t ws_size,
                              hipStream_t stream) {
    (void)in_sizes; (void)n_in; (void)out_size; (void)d_ws; (void)ws_size;
    const float* audio = (const float*)d_in[0];
    float* out = (float*)d_out;

    // audio-EQ-cookbook highpass coefficients, double math -> float32 (matches reference)
    const double PI = 3.14159265358979323846;
    const double cutoff = 700.0 / 16000.0;
    const double Q = 0.70710678;
    const double w0 = 2.0 * PI * cutoff;
    const double cw = cos(w0), sw = sin(w0);
    const double alpha = sw / (2.0 * Q);
    const double a0 = 1.0 + alpha;
    const float fb0 = (float)(((1.0 + cw) / 2.0) / a0);
    const float fb1 = (float)((-(1.0 + cw)) / a0);
    const float fb2 = (float)(((1.0 + cw) / 2.0) / a0);
    const float fa1 = (float)((-2.0 * cw) / a0);
    const float fa2 = (float)((1.0 - alpha) / a0);

    dim3 grid(NCH * 4);   // 256 channels x 4 waves (32 chunks each of 512 samples)
    dim3 block(32);       // one wave32 per workgroup
    hipLaunchKernelGGL(biquad_hp_kernel, grid, block, 0, stream,
                       audio, out, fb0, fb1, fb2, -fa1, -fa2);
}